// LatticeMaxPooling_16853451669851
// MI455X (gfx1250) — compile-verified
//
#include <hip/hip_runtime.h>
#include <stdint.h>
#include <math.h>

// Quincunx lattice max pooling, CDNA5 (gfx1250).
// out0[i,j] = max(c0[i,j], c0[i,j+1], c0[i+1,j], c0[i+1,j+1], c1[i,j])
// out1[i,j] = max(c1[i,j], c1[i,j+1], c1[i+1,j], c1[i+1,j+1], c0[i+1,j+1])

typedef float v4f __attribute__((ext_vector_type(4)));

constexpr int kH    = 512;
constexpr int kW    = 512;
constexpr int kTI   = 4;            // output rows per block (per coset)
constexpr int kLdsW = 516;          // padded LDS row stride in floats (2064 B, 16B-aligned)
constexpr int kRows = 2 * (kTI + 1); // 5 rows coset0 + 5 rows coset1

__device__ __forceinline__ unsigned lds_byte_off(const void* p) {
  // Low 32 bits of a flat shared-aperture address are the LDS byte offset.
  return (unsigned)(unsigned long long)(uintptr_t)p;
}

__device__ __forceinline__ v4f vmax4(v4f a, v4f b) {
  v4f r;
  r.x = fmaxf(a.x, b.x); r.y = fmaxf(a.y, b.y);
  r.z = fmaxf(a.z, b.z); r.w = fmaxf(a.w, b.w);
  return r;
}

__device__ __forceinline__ v4f shift1(v4f a, float e) {
  v4f r; r.x = a.y; r.y = a.z; r.z = a.w; r.w = e; return r;
}

__global__ __launch_bounds__(256) void quincunx_pool_kernel(
    const float* __restrict__ c0, const float* __restrict__ c1,
    float* __restrict__ out, int sliceCount) {
  __shared__ __align__(16) float smem[kRows * kLdsW];

  const int tid    = threadIdx.x;
  const int rowblk = blockIdx.x;                 // 0 .. H/kTI-1
  const int slice  = blockIdx.y;                 // 0 .. B*C-1
  const int i0     = rowblk * kTI;
  const size_t sliceOff = (size_t)slice * (kH * kW);

  // Column sentinel: element [kW] of each staged row = -inf (handles j+1 == W edge).
  if (tid < kRows) smem[tid * kLdsW + kW] = -INFINITY;

  // ---- Stage 5 rows of each coset via CDNA5 async global->LDS copies ----
  // 10 rows * 128 chunks(16B) = 1280 chunks; 256 threads * 5 iterations.
#pragma unroll
  for (int k = 0; k < 5; ++k) {
    int cid   = k * 256 + tid;
    int row   = cid >> 7;          // 0..9  (0..4 = coset0 rows, 5..9 = coset1 rows)
    int chunk = cid & 127;         // 16-byte chunk within the row
    int coset = (row >= 5) ? 1 : 0;
    int rr    = row - coset * 5;   // 0..4
    int srcRow = i0 + rr;
    if (srcRow > kH - 1) srcRow = kH - 1;  // clamp; predicated out in compute
    const float* gp =
        (coset ? c1 : c0) + sliceOff + (size_t)srcRow * kW + (size_t)chunk * 4;
    unsigned laddr = lds_byte_off(&smem[row * kLdsW + chunk * 4]);
    unsigned long long gaddr = (unsigned long long)(uintptr_t)gp;
    asm volatile("global_load_async_to_lds_b128 %0, %1, off"
                 :: "v"(laddr), "v"(gaddr)
                 : "memory");
  }
  // Wait for this wave's async copies, then cross-wave barrier.
  asm volatile("s_wait_asynccnt 0x0" ::: "memory");
  __syncthreads();

  const size_t outHalf = (size_t)sliceCount * (kH * kW); // coset-1 output offset

  // ---- Compute: 8 output rows (4 per coset) * 128 float4-chunks = 1024 chunks ----
#pragma unroll
  for (int k = 0; k < 4; ++k) {
    int oc   = k * 256 + tid;      // 0..1023
    int orow = oc >> 7;            // 0..7 (0..3 -> out0 rows, 4..7 -> out1 rows)
    int col  = (oc & 127) * 4;
    int t    = orow & 3;
    bool isOut1 = orow >= 4;
    int ownBase = isOut1 ? 5 : 0;
    bool rowValid = (i0 + t + 1) < kH;

    const float* pa = &smem[(ownBase + t) * kLdsW + col];      // own row i
    const float* pb = &smem[(ownBase + t + 1) * kLdsW + col];  // own row i+1
    v4f A = *(const v4f*)pa; float Ax = pa[4];
    v4f B = *(const v4f*)pb; float Bx = pb[4];

    v4f r  = vmax4(A, shift1(A, Ax));        // own[i,j], own[i,j+1]
    v4f nb = vmax4(B, shift1(B, Bx));        // own[i+1,j], own[i+1,j+1]

    if (isOut1) {
      // cross term c0[i+1, j+1] — gated by rowValid together with nb
      const float* pd = &smem[(t + 1) * kLdsW + col];
      v4f D = *(const v4f*)pd; float Dx = pd[4];
      nb = vmax4(nb, shift1(D, Dx));
    } else {
      // cross term c1[i, j] — always valid
      const float* pc = &smem[(5 + t) * kLdsW + col];
      v4f C = *(const v4f*)pc;
      r = vmax4(r, C);
    }
    if (rowValid) r = vmax4(r, nb);

    float* po = out + (isOut1 ? outHalf : (size_t)0) + sliceOff +
                (size_t)(i0 + t) * kW + col;
    __builtin_nontemporal_store(r, (v4f*)po);  // streaming store, keep L2 for inputs
  }
}

extern "C" void kernel_launch(void* const* d_in, const int* in_sizes, int n_in,
                              void* d_out, int out_size, void* d_ws, size_t ws_size,
                              hipStream_t stream) {
  const float* c0 = (const float*)d_in[0];
  const float* c1 = (const float*)d_in[1];
  float* out = (float*)d_out;
  int slices = in_sizes[0] / (kH * kW);   // B*C = 128
  dim3 grid(kH / kTI, slices);
  quincunx_pool_kernel<<<grid, 256, 0, stream>>>(c0, c1, out, slices);
}